// LossWithBeliveMaps_22789096472576
// MI455X (gfx1250) — compile-verified
//
#include <hip/hip_runtime.h>
#include <stdint.h>

// Problem constants (from reference setup_inputs)
#define BQ 8
#define SQ 8
#define NQ 32
#define HQ 512
#define WQ 512
#define HWQ (HQ * WQ)          // 262144 pixels per (b,s) slice
#define KK 9
#define PIXPERBLK 1024         // 256 threads * 4 floats
#define NBLK_MAIN (BQ * HWQ / PIXPERBLK)   // 2048

typedef __attribute__((ext_vector_type(2))) float v2f;
typedef __attribute__((ext_vector_type(8))) float v8f;

// ---------------------------------------------------------------------------
// CDNA5 async global->LDS DMA (ASYNCcnt-tracked). GVS addressing:
//   LDS byte address from a VGPR; global address = 64-bit SGPR base + 32-bit VGPR offset
// ---------------------------------------------------------------------------
__device__ __forceinline__ void async_load_b128_to_lds(uint32_t lds_byte_off,
                                                       uint32_t glb_byte_off,
                                                       const void* sbase) {
  asm volatile("global_load_async_to_lds_b128 %0, %1, %2"
               :: "v"(lds_byte_off), "v"(glb_byte_off), "s"(sbase)
               : "memory");
}

__device__ __forceinline__ void wait_async_le(int n) {
  switch (n) {   // constant-folds under #pragma unroll
    case 0: asm volatile("s_wait_asynccnt 0" ::: "memory"); break;
    case 1: asm volatile("s_wait_asynccnt 1" ::: "memory"); break;
    case 2: asm volatile("s_wait_asynccnt 2" ::: "memory"); break;
    case 3: asm volatile("s_wait_asynccnt 3" ::: "memory"); break;
    case 4: asm volatile("s_wait_asynccnt 4" ::: "memory"); break;
    case 5: asm volatile("s_wait_asynccnt 5" ::: "memory"); break;
    case 6: asm volatile("s_wait_asynccnt 6" ::: "memory"); break;
    default: asm volatile("s_wait_asynccnt 7" ::: "memory"); break;
  }
}

// ---------------------------------------------------------------------------
// Block-wide sum via the matrix unit: V_WMMA_F32_16X16X4_F32 with B = ones.
//   A(16x4) holds 64 values (lane's {v, 0} pair), D[m][n] = sum_k A[m][k].
//   C/D layout: lane n holds D[0..7][n] in its 8 acc regs; lane n+16 holds
//   D[8..15][n]  ->  wave sum = (sum of 8 regs) + shfl_xor(.,16).
// Requires EXEC all-ones (full 256-thread block, no divergence here).
// Valid result returned on tid==0 only.
// ---------------------------------------------------------------------------
__device__ __forceinline__ float block_reduce_sum_wmma(float v, float* redw /*__shared__[8]*/) {
  v2f a; a.x = v;   a.y = 0.f;
  v2f b; b.x = 1.f; b.y = 1.f;
  v8f c = {};
  v8f d = __builtin_amdgcn_wmma_f32_16x16x4_f32(
      /*neg_a=*/false, a, /*neg_b=*/false, b,
      /*c_mod=*/(short)0, c, /*reuse_a=*/false, /*reuse_b=*/false);
  float t = ((d[0] + d[1]) + (d[2] + d[3])) + ((d[4] + d[5]) + (d[6] + d[7]));
  t += __shfl_xor(t, 16, 32);           // combine M=0..7 half with M=8..15 half
  const int tid = threadIdx.x;
  if ((tid & 31) == 0) redw[tid >> 5] = t;
  __syncthreads();
  float total = 0.f;
  if (tid == 0) {
    #pragma unroll
    for (int i = 0; i < 8; ++i) total += redw[i];
  }
  return total;
}

// ---------------------------------------------------------------------------
// Kernel 0: zero dense belive_map (8 MiB) + f64 accumulator
// ---------------------------------------------------------------------------
__global__ void k_zero(float4* __restrict__ bmap4, double* __restrict__ acc) {
  int i = blockIdx.x * 256 + threadIdx.x;      // [0, BQ*HWQ/4)
  bmap4[i] = float4{0.f, 0.f, 0.f, 0.f};
  if (i == 0) *acc = 0.0;
}

// ---------------------------------------------------------------------------
// Kernel 1: sparse "conv" — dedupe keypoints (matches .set(1.0) semantics)
// then scatter-add the 9x9 Gaussian patch with border clipping.
// ---------------------------------------------------------------------------
__global__ void k_scatter(const int* __restrict__ tgt,
                          const float* __restrict__ kern,
                          float* __restrict__ bmap) {
  int t = blockIdx.x * blockDim.x + threadIdx.x;
  if (t >= BQ * NQ * SQ) return;
  int b = t >> 8;
  int n = (t >> 3) & (NQ - 1);
  int s = t & (SQ - 1);
  int base = ((b * NQ + n) * SQ + s) * 2;
  int x = tgt[base + 0];
  int y = tgt[base + 1];
  for (int n2 = 0; n2 < n; ++n2) {
    int b2 = ((b * NQ + n2) * SQ + s) * 2;
    if (tgt[b2] == x && tgt[b2 + 1] == y) return;
  }
  float* bimg = bmap + (size_t)b * HWQ;
  #pragma unroll
  for (int dy = 0; dy < KK; ++dy) {
    int h = y - 4 + dy;
    if ((unsigned)h >= (unsigned)HQ) continue;
    #pragma unroll
    for (int dx = 0; dx < KK; ++dx) {
      int w = x - 4 + dx;
      if ((unsigned)w >= (unsigned)WQ) continue;
      atomicAdd(&bimg[h * WQ + w], kern[dy * KK + dx]); // kernel is symmetric
    }
  }
}

// ---------------------------------------------------------------------------
// Kernel 2 (main, bandwidth-bound): each block = 1024 contiguous pixels of
// one image b. 8-stage async global->LDS pipeline streams the 8 S-channels;
// VALU drains LDS with FMA accumulation of (pred - bmap)^2; WMMA reduction.
// ---------------------------------------------------------------------------
__global__ void k_main_fast(const float* __restrict__ pred,
                            const float* __restrict__ bmap,
                            double* __restrict__ acc) {
  __shared__ __align__(16) float abuf[SQ][PIXPERBLK];  // 32 KiB pipeline
  __shared__ float redw[8];

  const int tid   = threadIdx.x;
  const int b     = blockIdx.x >> 8;            // 256 blocks per image
  const int chunk = blockIdx.x & 255;
  const int pix   = chunk * PIXPERBLK + tid * 4; // pixel index inside image
  const uint32_t voff = (uint32_t)pix * 4u;      // byte offset in (b,s) slice

  // Issue the full 8-deep async DMA pipeline (ASYNCcnt = 8 outstanding).
  #pragma unroll
  for (int s = 0; s < SQ; ++s) {
    const float* sbase = pred + (size_t)(b * SQ + s) * HWQ;
    uint32_t la = (uint32_t)(uintptr_t)(&abuf[s][tid * 4]);
    async_load_b128_to_lds(la, voff, sbase);
  }

  // belive_map tile: one b128 load, lives in registers for all 8 channels.
  const float4 bm = *(const float4*)(bmap + (size_t)b * HWQ + pix);

  float a0 = 0.f, a1 = 0.f, a2 = 0.f, a3 = 0.f;
  #pragma unroll
  for (int s = 0; s < SQ; ++s) {
    wait_async_le(SQ - 1 - s);   // async loads complete in order
    float4 p = *(const float4*)(&abuf[s][tid * 4]);  // ds_load_b128
    float d0 = p.x - bm.x, d1 = p.y - bm.y, d2 = p.z - bm.z, d3 = p.w - bm.w;
    a0 = fmaf(d0, d0, a0);
    a1 = fmaf(d1, d1, a1);
    a2 = fmaf(d2, d2, a2);
    a3 = fmaf(d3, d3, a3);
  }

  float total = block_reduce_sum_wmma((a0 + a1) + (a2 + a3), redw);
  if (tid == 0) atomicAdd(acc, (double)total);
}

// ---------------------------------------------------------------------------
// Kernel 3: finalize mean
// ---------------------------------------------------------------------------
__global__ void k_final(const double* __restrict__ acc, float* __restrict__ out) {
  out[0] = (float)(*acc / (double)((size_t)BQ * SQ * HWQ));
}

// ---------------------------------------------------------------------------
// Fallback path (workspace too small for dense bmap): flags + on-the-fly bmap
// ---------------------------------------------------------------------------
__global__ void k_flags(const int* __restrict__ tgt, unsigned char* __restrict__ flags,
                        double* __restrict__ acc) {
  int t = blockIdx.x * blockDim.x + threadIdx.x;
  if (t >= BQ * NQ * SQ) return;
  if (t == 0) *acc = 0.0;
  int b = t >> 8, n = (t >> 3) & (NQ - 1), s = t & (SQ - 1);
  int base = ((b * NQ + n) * SQ + s) * 2;
  int x = tgt[base], y = tgt[base + 1];
  unsigned char keep = 1;
  for (int n2 = 0; n2 < n; ++n2) {
    int b2 = ((b * NQ + n2) * SQ + s) * 2;
    if (tgt[b2] == x && tgt[b2 + 1] == y) { keep = 0; break; }
  }
  flags[t] = keep;
}

__global__ void k_main_slow(const float* __restrict__ pred,
                            const int* __restrict__ tgt,
                            const float* __restrict__ kern,
                            const unsigned char* __restrict__ flags,
                            double* __restrict__ acc) {
  __shared__ float redw[8];
  int tid = threadIdx.x;
  size_t g = (size_t)blockIdx.x * 256 + tid;   // over B*H*W
  int b = (int)(g / HWQ);
  int pix = (int)(g % HWQ);
  int h = pix >> 9, w = pix & (WQ - 1);
  float bm = 0.f;
  for (int t = 0; t < NQ * SQ; ++t) {
    if (!flags[b * NQ * SQ + t]) continue;
    int base = ((b * NQ + (t >> 3)) * SQ + (t & 7)) * 2;
    int x = tgt[base], y = tgt[base + 1];
    int iy = h - y + 4, ix = w - x + 4;
    if ((unsigned)iy < (unsigned)KK && (unsigned)ix < (unsigned)KK)
      bm += kern[iy * KK + ix];
  }
  float a = 0.f;
  #pragma unroll
  for (int s = 0; s < SQ; ++s) {
    float d = pred[(size_t)(b * SQ + s) * HWQ + pix] - bm;
    a = fmaf(d, d, a);
  }
  float total = block_reduce_sum_wmma(a, redw);
  if (tid == 0) atomicAdd(acc, (double)total);
}

// ---------------------------------------------------------------------------
extern "C" void kernel_launch(void* const* d_in, const int* in_sizes, int n_in,
                              void* d_out, int out_size, void* d_ws, size_t ws_size,
                              hipStream_t stream) {
  (void)in_sizes; (void)n_in; (void)out_size;
  const float* pred = (const float*)d_in[0];   // (B,S,H,W) fp32
  const int*   tgt  = (const int*)d_in[1];     // (B,N,S,2) int32
  const float* kern = (const float*)d_in[2];   // (9,9) fp32
  float* out = (float*)d_out;

  double* acc = (double*)d_ws;
  unsigned char* flags = (unsigned char*)d_ws + 64;
  float* bmap = (float*)((char*)d_ws + 8192);

  const size_t need_fast = 8192 + (size_t)BQ * HWQ * sizeof(float);
  if (ws_size >= need_fast) {
    k_zero<<<NBLK_MAIN, 256, 0, stream>>>((float4*)bmap, acc);
    k_scatter<<<(BQ * NQ * SQ + 255) / 256, 256, 0, stream>>>(tgt, kern, bmap);
    k_main_fast<<<NBLK_MAIN, 256, 0, stream>>>(pred, bmap, acc);
  } else {
    k_flags<<<(BQ * NQ * SQ + 255) / 256, 256, 0, stream>>>(tgt, flags, acc);
    k_main_slow<<<(BQ * HWQ) / 256, 256, 0, stream>>>(pred, tgt, kern, flags, acc);
  }
  k_final<<<1, 1, 0, stream>>>(acc, out);
}